// Encoder_76819785056520
// MI455X (gfx1250) — compile-verified
//
#include <hip/hip_runtime.h>

typedef float v2f __attribute__((ext_vector_type(2)));
typedef float v8f __attribute__((ext_vector_type(8)));

#define DIN 512
#define H1F 128
#define H2F 64

// ---------------- CSR build ----------------
__global__ void k_zero_i32(int* p, int n) {
    int i = blockIdx.x * blockDim.x + threadIdx.x;
    if (i < n) p[i] = 0;
}

__global__ void k_count(const int* __restrict__ dst, int* __restrict__ cnt, int E) {
    int e = blockIdx.x * blockDim.x + threadIdx.x;
    if (e < E) atomicAdd(&cnt[dst[e]], 1);
}

// inclusive block scan (Hillis-Steele), 1024 per block
__global__ void k_scan_block(const int* __restrict__ cnt, int* __restrict__ incl,
                             int* __restrict__ bsum, int n) {
    __shared__ int s[1024];
    int t = threadIdx.x, i = blockIdx.x * 1024 + t;
    s[t] = (i < n) ? cnt[i] : 0;
    __syncthreads();
    for (int d = 1; d < 1024; d <<= 1) {
        int x = (t >= d) ? s[t - d] : 0;
        __syncthreads();
        s[t] += x;
        __syncthreads();
    }
    if (i < n) incl[i] = s[t];
    if (t == 1023) bsum[blockIdx.x] = s[t];
}

__global__ void k_scan_sums(int* __restrict__ bsum, int nb) {
    __shared__ int s[1024];
    int t = threadIdx.x;
    s[t] = (t < nb) ? bsum[t] : 0;
    __syncthreads();
    for (int d = 1; d < 1024; d <<= 1) {
        int x = (t >= d) ? s[t - d] : 0;
        __syncthreads();
        s[t] += x;
        __syncthreads();
    }
    if (t < nb) bsum[t] = s[t];
}

// turn inclusive scan into exclusive rowptr; init cursor; dinv = rsqrt(deg) with self-loop
__global__ void k_finalize(const int* __restrict__ cnt, int* __restrict__ rowptr,
                           const int* __restrict__ bsum, int* __restrict__ cursor,
                           float* __restrict__ dinv, int n) {
    int i = blockIdx.x * 1024 + threadIdx.x;
    if (i >= n) return;
    int off  = blockIdx.x ? bsum[blockIdx.x - 1] : 0;
    int incl = rowptr[i] + off;
    int excl = incl - cnt[i];
    rowptr[i] = excl;
    cursor[i] = excl;
    dinv[i] = rsqrtf((float)(cnt[i] + 1));   // +1 self-loop; always > 0
}

__global__ void k_fill(const int* __restrict__ src, const int* __restrict__ dst,
                       int* __restrict__ cursor, int* __restrict__ csr, int E) {
    int e = blockIdx.x * blockDim.x + threadIdx.x;
    if (e < E) {
        int p = atomicAdd(&cursor[dst[e]], 1);
        csr[p] = src[e];
    }
}

// ---------------- FP32 WMMA GEMM: C[M x NOUT] = A[M x K] * B[K x NOUT] ----------------
// One block per 16-row M tile; one wave per 16-col N tile. M % 16 == 0 (EXEC all-ones for WMMA).
template <int K, int NOUT>
__global__ __launch_bounds__(32 * (NOUT / 16))
void k_gemm_wmma(const float* __restrict__ A, const float* __restrict__ B,
                 float* __restrict__ C) {
    constexpr int NT = NOUT / 16;
    constexpr int KP = K + 4;               // K+4 % 64 == 4 -> 16 rows hit 16 distinct LDS banks
    __shared__ float As[16 * KP];

    const int tid   = threadIdx.x;
    const int mbase = blockIdx.x * 16;

    // cooperative float4 fill of the 16 x K A-tile
    constexpr int NV = 16 * K / 4;
    for (int i = tid; i < NV; i += 32 * NT) {
        int r  = i / (K / 4);
        int c4 = i - r * (K / 4);
        float4 v = *reinterpret_cast<const float4*>(A + (size_t)(mbase + r) * K + 4 * c4);
        *reinterpret_cast<float4*>(&As[r * KP + 4 * c4]) = v;
    }
    __syncthreads();

    const int wave  = tid >> 5;
    const int lane  = tid & 31;
    const int lh    = lane & 15;            // row (A) / col (B,D) within tile
    const int kh    = (lane >> 4) << 1;     // K sub-offset: 0 (lanes 0-15) or 2 (lanes 16-31)
    const int nbase = wave * 16;

    v8f acc = {0.f, 0.f, 0.f, 0.f, 0.f, 0.f, 0.f, 0.f};
    const float* as_row = &As[lh * KP + kh];
    const float* bp     = B + (size_t)kh * NOUT + nbase + lh;

#pragma unroll 4
    for (int k = 0; k < K; k += 4) {
        v2f a = *reinterpret_cast<const v2f*>(as_row + k);   // ds_load_b64
        v2f b;
        b.x = bp[(size_t)k * NOUT];
        b.y = bp[(size_t)(k + 1) * NOUT];
        acc = __builtin_amdgcn_wmma_f32_16x16x4_f32(false, a, false, b,
                                                    (short)0, acc, false, false);
    }

    // D layout: VGPR g holds row g (lanes 0-15) / row g+8 (lanes 16-31), col = lane%16
#pragma unroll
    for (int g = 0; g < 8; ++g) {
        int row = mbase + g + ((lane >> 4) << 3);
        C[(size_t)row * NOUT + nbase + lh] = acc[g];
    }
}

// ---------------- per-node CSR aggregation: out = relu(sum_e norm*hw[src] + self + bias) ----------------
// one wave per node; lane owns VPT consecutive features (F = 32*VPT)
template <int VPT>
__global__ void k_agg(const float* __restrict__ hw, const int* __restrict__ rowptr,
                      const int* __restrict__ cnt, const int* __restrict__ csr,
                      const float* __restrict__ dinv, const float* __restrict__ bias,
                      float* __restrict__ out, int n) {
    constexpr int F = 32 * VPT;
    int wave = threadIdx.x >> 5;
    int lane = threadIdx.x & 31;
    int node = blockIdx.x * (blockDim.x >> 5) + wave;
    if (node >= n) return;

    float di = dinv[node];
    float acc[VPT];

    const float* selfp = hw + (size_t)node * F + lane * VPT;
    if constexpr (VPT == 4) {
        float4 v = *reinterpret_cast<const float4*>(selfp);
        acc[0] = v.x * di * di; acc[1] = v.y * di * di;
        acc[2] = v.z * di * di; acc[3] = v.w * di * di;
    } else {
        float2 v = *reinterpret_cast<const float2*>(selfp);
        acc[0] = v.x * di * di; acc[1] = v.y * di * di;
    }

    int beg = rowptr[node];
    int c   = cnt[node];
    for (int i = 0; i < c; ++i) {
        int s   = csr[beg + i];                     // scalar (uniform per wave)
        float w = dinv[s] * di;
        const float* p = hw + (size_t)s * F + lane * VPT;
        if (i + 1 < c)
            __builtin_prefetch(hw + (size_t)csr[beg + i + 1] * F + lane * VPT, 0, 0);
        if constexpr (VPT == 4) {
            float4 v = *reinterpret_cast<const float4*>(p);
            acc[0] += v.x * w; acc[1] += v.y * w;
            acc[2] += v.z * w; acc[3] += v.w * w;
        } else {
            float2 v = *reinterpret_cast<const float2*>(p);
            acc[0] += v.x * w; acc[1] += v.y * w;
        }
    }

    float* o = out + (size_t)node * F + lane * VPT;
#pragma unroll
    for (int j = 0; j < VPT; ++j) {
        float v = acc[j] + bias[lane * VPT + j];
        o[j] = v > 0.f ? v : 0.f;
    }
}

// ---------------- driver ----------------
extern "C" void kernel_launch(void* const* d_in, const int* in_sizes, int n_in,
                              void* d_out, int out_size, void* d_ws, size_t ws_size,
                              hipStream_t stream) {
    const float* x  = (const float*)d_in[0];
    const int*   ei = (const int*)d_in[1];
    const float* W1 = (const float*)d_in[2];
    const float* b1 = (const float*)d_in[3];
    const float* W2 = (const float*)d_in[4];
    const float* b2 = (const float*)d_in[5];

    const int n = in_sizes[0] / DIN;     // 50000 (multiple of 16)
    const int E = in_sizes[1] / 2;       // 800000
    const int* srcI = ei;
    const int* dstI = ei + E;

    char* ws = (char*)d_ws;
    size_t off = 0;
    auto alloc = [&](size_t bytes) -> void* {
        void* p = ws + off;
        off = (off + bytes + 255) & ~(size_t)255;
        return p;
    };
    int nb = (n + 1023) / 1024;
    int*   cnt    = (int*)alloc((size_t)n * 4);
    int*   rowptr = (int*)alloc((size_t)n * 4);
    int*   cursor = (int*)alloc((size_t)n * 4);
    int*   bsum   = (int*)alloc((size_t)nb * 4);
    float* dinv   = (float*)alloc((size_t)n * 4);
    int*   csr    = (int*)alloc((size_t)E * 4);
    float* hw1    = (float*)alloc((size_t)n * H1F * 4);
    float* h      = (float*)alloc((size_t)n * H1F * 4);
    float* hw2    = hw1;                 // hw1 dead after layer-1 aggregation
    float* outp   = (float*)d_out;

    // CSR by destination (shared by both layers) + symmetric normalization
    k_zero_i32<<<(n + 255) / 256, 256, 0, stream>>>(cnt, n);
    k_count<<<(E + 255) / 256, 256, 0, stream>>>(dstI, cnt, E);
    k_scan_block<<<nb, 1024, 0, stream>>>(cnt, rowptr, bsum, n);
    k_scan_sums<<<1, 1024, 0, stream>>>(bsum, nb);
    k_finalize<<<nb, 1024, 0, stream>>>(cnt, rowptr, bsum, cursor, dinv, n);
    k_fill<<<(E + 255) / 256, 256, 0, stream>>>(srcI, dstI, cursor, csr, E);

    // layer 1: hw1 = x @ W1 (WMMA), then conflict-free CSR aggregation + bias + relu
    k_gemm_wmma<DIN, H1F><<<n / 16, 32 * (H1F / 16), 0, stream>>>(x, W1, hw1);
    k_agg<4><<<(n + 7) / 8, 256, 0, stream>>>(hw1, rowptr, cnt, csr, dinv, b1, h, n);

    // layer 2
    k_gemm_wmma<H1F, H2F><<<n / 16, 32 * (H2F / 16), 0, stream>>>(h, W2, hw2);
    k_agg<2><<<(n + 7) / 8, 256, 0, stream>>>(hw2, rowptr, cnt, csr, dinv, b2, outp, n);
}